// QCNN_65481071399202
// MI455X (gfx1250) — compile-verified
//
#include <hip/hip_runtime.h>
#include <math.h>

typedef __attribute__((ext_vector_type(2))) float v2f;
typedef __attribute__((ext_vector_type(8))) float v8f;

#define NW 10
#define TPB 256   // one thread per batch element; 8 waves per block

__global__ __launch_bounds__(TPB)
void qcnn_kernel(const float* __restrict__ inputs,   // (B, 10)
                 const float* __restrict__ thetas,   // (6, 10)
                 const float* __restrict__ Wm,       // (10, 4)
                 const float* __restrict__ bias,     // (4,)
                 float* __restrict__ out)            // (B, 4)
{
    __shared__ float lX[TPB][12];   // input staging, then feature rows (K padded 10->12)
    __shared__ float lBt[16][12];   // W transposed+zero-padded: lBt[n][k] = W[k][n]
    __shared__ float lBias[16];     // bias zero-padded to 16
    __shared__ float lO[TPB][4];    // sigmoid outputs for coalesced store

    const int tid       = threadIdx.x;
    const int blockBase = blockIdx.x * TPB;

    // ---- stage: coalesced inputs (10x 1KB-contiguous loads) + padded weights ----
    #pragma unroll
    for (int j = 0; j < NW; ++j) {
        const int idx = j * TPB + tid;                 // 0..2559, fully coalesced
        lX[idx / NW][idx % NW] = inputs[blockBase * NW + idx];
    }
    if (tid < 192) {                                    // 16 cols x 12 rows of B^T
        const int n = tid / 12, k = tid % 12;
        const bool valid = (k < NW) && (n < 4);
        const float v = Wm[valid ? k * 4 + n : 0];      // unconditional load, clamped addr
        lBt[n][k] = valid ? v : 0.0f;
    }
    if (tid < 16) {
        const float v = bias[tid & 3];
        lBias[tid] = (tid < 4) ? v : 0.0f;
    }
    __syncthreads();

    // ---- closed-form circuit: x <- cumprod_w cos(x_w + theta_k_w), 6 layers ----
    // (RX gates commute across wires and compose per wire; the CNOT chain folds
    //  into parity signs; product-state expectations factorize to cumprod(cos).)
    float x[NW];
    #pragma unroll
    for (int w = 0; w < NW; ++w)
        x[w] = lX[tid][w];

    #pragma unroll
    for (int k = 0; k < 6; ++k) {
        float cum = 1.0f;
        #pragma unroll
        for (int w = 0; w < NW; ++w) {
            cum *= cosf(x[w] + thetas[k * NW + w]);    // thetas: uniform scalar loads
            x[w] = cum;
        }
    }

    // own row was read only by this thread -> safe to overwrite without a barrier
    #pragma unroll
    for (int w = 0; w < 12; ++w)
        lX[tid][w] = (w < NW) ? x[w] : 0.0f;
    __syncthreads();

    // ---- final linear on the matrix core: A(16x12) @ B(12x16, W padded) via
    //      3 chained V_WMMA_F32_16X16X4_F32; each wave does its 32 rows as 2 tiles ----
    const int wave = tid >> 5;
    const int lane = tid & 31;
    const int n    = lane & 15;            // A-row (M) / B-col (N) index
    const int ks   = (lane >> 4) << 1;     // K sub-offset: lanes 0-15 -> {0,1}, 16-31 -> {2,3}
    const float bj = lBias[n];

    #pragma unroll
    for (int half = 0; half < 2; ++half) {
        const int rowbase = wave * 32 + half * 16;

        v8f cfrag = {};
        #pragma unroll
        for (int t = 0; t < 3; ++t) {
            const int k0 = 4 * t + ks;
            const v2f af = *(const v2f*)&lX[rowbase + n][k0];   // ds_load_b64
            const v2f bf = *(const v2f*)&lBt[n][k0];            // ds_load_b64
            cfrag = __builtin_amdgcn_wmma_f32_16x16x4_f32(
                false, af, false, bf, (short)0, cfrag, false, false);
        }

        #pragma unroll
        for (int v = 0; v < 8; ++v) {
            const int M = v + ((lane < 16) ? 0 : 8);   // D layout: vgpr v -> M = v (+8 hi lanes)
            if (n < 4)
                lO[rowbase + M][n] = 1.0f / (1.0f + expf(-(cfrag[v] + bj)));
        }
    }
    __syncthreads();

    // ---- coalesced 128-bit store: one float4 per batch row ----
    ((float4*)out)[blockBase + tid] = *(const float4*)lO[tid];
}

extern "C" void kernel_launch(void* const* d_in, const int* in_sizes, int n_in,
                              void* d_out, int out_size, void* d_ws, size_t ws_size,
                              hipStream_t stream) {
    const float* inputs = (const float*)d_in[0];
    const float* thetas = (const float*)d_in[1];
    const float* Wm     = (const float*)d_in[2];
    const float* bias   = (const float*)d_in[3];
    float* out = (float*)d_out;

    const int B = in_sizes[0] / NW;          // 32768
    const int blocks = B / TPB;              // 128 blocks x 256 threads
    qcnn_kernel<<<blocks, TPB, 0, stream>>>(inputs, thetas, Wm, bias, out);
}